// DeQuConvXYZ_6794638263036
// MI455X (gfx1250) — compile-verified
//
#include <hip/hip_runtime.h>
#include <hip/hip_bf16.h>

// ---------------------------------------------------------------------------
// out = E X E^H,  E = (u2 kron^6)^dagger.  Let F = u2^{kron 6}. Then
//   out = F^H X F
//   Tr = X Fr ; Ti = X Fi
//   outr = Fr^T Tr + Fi^T Ti ; outi = Fr^T Ti - Fi^T Tr
// 6 real fp32 4096^3 GEMMs with V_WMMA_F32_16X16X4_F32 (wave32),
// tiles staged by the Tensor Data Mover (TENSOR_LOAD_TO_LDS) with
// double-buffered LDS so DMA overlaps WMMA.
// ---------------------------------------------------------------------------

typedef __attribute__((ext_vector_type(2))) float v2f;
typedef __attribute__((ext_vector_type(8))) float v8f;
typedef __attribute__((ext_vector_type(4))) unsigned int v4u;
typedef __attribute__((ext_vector_type(8))) int v8i;
typedef __attribute__((ext_vector_type(4))) int v4i;

constexpr int NMAT = 4096;
constexpr int BM = 128;
constexpr int BN = 64;
constexpr int BK = 16;

struct Cpx { float r, i; };

__device__ inline Cpx cmul(Cpx a, Cpx b) {
    return { a.r * b.r - a.i * b.i, a.r * b.i + a.i * b.r };
}
__device__ inline Cpx cadd(Cpx a, Cpx b) { return { a.r + b.r, a.i + b.i }; }

__device__ inline void zero4(Cpx G[4][4]) {
    for (int m = 0; m < 4; ++m)
        for (int n = 0; n < 4; ++n) G[m][n] = {0.f, 0.f};
}

__device__ inline void kron_R_I(Cpx G[4][4], const Cpx R[2][2]) {
    zero4(G);
    for (int a = 0; a < 2; ++a)
        for (int b = 0; b < 2; ++b)
            for (int r = 0; r < 2; ++r)
                G[2 * a + r][2 * b + r] = R[a][b];
}
__device__ inline void kron_I_R(Cpx G[4][4], const Cpx R[2][2]) {
    zero4(G);
    for (int r = 0; r < 2; ++r)
        for (int a = 0; a < 2; ++a)
            for (int b = 0; b < 2; ++b)
                G[2 * r + a][2 * r + b] = R[a][b];
}

// U = G @ U
__device__ inline void lmul4(Cpx U[4][4], const Cpx G[4][4]) {
    Cpx R[4][4];
    for (int m = 0; m < 4; ++m)
        for (int n = 0; n < 4; ++n) {
            Cpx s = {0.f, 0.f};
            for (int k = 0; k < 4; ++k) s = cadd(s, cmul(G[m][k], U[k][n]));
            R[m][n] = s;
        }
    for (int m = 0; m < 4; ++m)
        for (int n = 0; n < 4; ++n) U[m][n] = R[m][n];
}

__device__ inline void rot_rx(Cpx R[2][2], float t) {
    float c = __cosf(t * 0.5f), s = __sinf(t * 0.5f);
    R[0][0] = {c, 0.f}; R[0][1] = {0.f, -s};
    R[1][0] = {0.f, -s}; R[1][1] = {c, 0.f};
}
__device__ inline void rot_ry(Cpx R[2][2], float t) {
    float c = __cosf(t * 0.5f), s = __sinf(t * 0.5f);
    R[0][0] = {c, 0.f}; R[0][1] = {-s, 0.f};
    R[1][0] = {s, 0.f}; R[1][1] = {c, 0.f};
}
__device__ inline void rot_rz(Cpx R[2][2], float t) {
    float c = __cosf(t * 0.5f), s = __sinf(t * 0.5f);
    R[0][0] = {c, -s}; R[0][1] = {0.f, 0.f};
    R[1][0] = {0.f, 0.f}; R[1][1] = {c, s};
}

__global__ void build_u2_kernel(const float* __restrict__ weight,
                                float* __restrict__ u2out) {
    if (threadIdx.x != 0 || blockIdx.x != 0) return;
    const float W_MUL = 0.632455532033675866f;  // sqrt(2/5)
    float w[15];
    for (int i = 0; i < 15; ++i) w[i] = weight[i] * W_MUL;

    Cpx U[4][4];
    zero4(U);
    for (int d = 0; d < 4; ++d) U[d][d] = {1.f, 0.f};

    Cpx R[2][2], G[4][4];

    rot_rx(R, w[0]);  kron_R_I(G, R); lmul4(U, G);
    rot_rx(R, w[1]);  kron_I_R(G, R); lmul4(U, G);
    rot_ry(R, w[2]);  kron_R_I(G, R); lmul4(U, G);
    rot_ry(R, w[3]);  kron_I_R(G, R); lmul4(U, G);
    rot_rz(R, w[4]);  kron_R_I(G, R); lmul4(U, G);
    rot_rz(R, w[5]);  kron_I_R(G, R); lmul4(U, G);

    {   // rxx
        float c = __cosf(w[6] * 0.5f), s = __sinf(w[6] * 0.5f);
        zero4(G);
        for (int d = 0; d < 4; ++d) G[d][d] = {c, 0.f};
        G[0][3] = {0.f, -s}; G[1][2] = {0.f, -s};
        G[2][1] = {0.f, -s}; G[3][0] = {0.f, -s};
        lmul4(U, G);
    }
    {   // ryy
        float c = __cosf(w[7] * 0.5f), s = __sinf(w[7] * 0.5f);
        zero4(G);
        for (int d = 0; d < 4; ++d) G[d][d] = {c, 0.f};
        G[0][3] = {0.f,  s}; G[1][2] = {0.f, -s};
        G[2][1] = {0.f, -s}; G[3][0] = {0.f,  s};
        lmul4(U, G);
    }
    {   // rzz: diag(exp(-i t/2 * sign)), signs {1,-1,-1,1}
        float c = __cosf(w[8] * 0.5f), s = __sinf(w[8] * 0.5f);
        const float sgn[4] = {1.f, -1.f, -1.f, 1.f};
        zero4(G);
        for (int d = 0; d < 4; ++d) G[d][d] = {c, -sgn[d] * s};
        lmul4(U, G);
    }

    rot_rx(R, w[9]);   kron_R_I(G, R); lmul4(U, G);
    rot_rx(R, w[10]);  kron_I_R(G, R); lmul4(U, G);
    rot_ry(R, w[11]);  kron_R_I(G, R); lmul4(U, G);
    rot_ry(R, w[12]);  kron_I_R(G, R); lmul4(U, G);
    rot_rz(R, w[13]);  kron_R_I(G, R); lmul4(U, G);
    rot_rz(R, w[14]);  kron_I_R(G, R); lmul4(U, G);

    for (int m = 0; m < 4; ++m)
        for (int n = 0; n < 4; ++n) {
            u2out[m * 4 + n]      = U[m][n].r;
            u2out[16 + m * 4 + n] = U[m][n].i;
        }
}

// F[i,j] = prod over the 6 base-4 digit pairs of u2[i_k][j_k]
__global__ void build_F_kernel(const float* __restrict__ u2,
                               float* __restrict__ Fr,
                               float* __restrict__ Fi) {
    __shared__ float su[32];
    if (threadIdx.x < 32) su[threadIdx.x] = u2[threadIdx.x];
    __syncthreads();
    size_t idx = (size_t)blockIdx.x * 256 + threadIdx.x;
    int i = (int)(idx >> 12);
    int j = (int)(idx & 4095);
    float pr = 1.f, pi = 0.f;
    for (int k = 0; k < 6; ++k) {
        int di = (i >> (2 * k)) & 3;
        int dj = (j >> (2 * k)) & 3;
        float gr = su[di * 4 + dj];
        float gi = su[16 + di * 4 + dj];
        float nr = pr * gr - pi * gi;
        float ni = pr * gi + pi * gr;
        pr = nr; pi = ni;
    }
    Fr[idx] = pr;
    Fi[idx] = pi;
}

// ---------------------------------------------------------------------------
// TDM: issue one 2-D tile load (tile_w x tile_h dwords, row stride in dwords)
// into LDS at lds_addr, inserting pad dwords after every row per the encoded
// pad_interval/pad_amount codes (ISA D# group1 fields).
// D# packing per CDNA5 ISA.
// Toolchain: clang-23 6-arg builtin form
//   (uint32x4 g0, int32x8 g1, int32x4, int32x4, int32x8, i32 cpol)
// ---------------------------------------------------------------------------
__device__ __attribute__((always_inline))
inline void tdm_load_2d(unsigned lds_addr, const float* gptr,
                        int tile_w, int tile_h, int row_stride,
                        int pad_interval_code, int pad_amount_code) {
    unsigned long long ga = (unsigned long long)(uintptr_t)gptr;

    v4u g0;
    g0[0] = 1u;                                            // count=1, user mode
    g0[1] = lds_addr;                                      // lds_addr (bytes)
    g0[2] = (unsigned)(ga & 0xFFFFFFFFu);                  // global_addr[31:0]
    g0[3] = (unsigned)((ga >> 32) & 0x1FFFFFFu) | (2u << 30); // [56:32] | type=2

    const unsigned dim0 = 1u << 20;   // tensor dims >> tile dims: no OOB clip
    const unsigned dim1 = 1u << 20;

    v8i g1;
    g1[0] = (int)((2u << 16)                               // data_size = 4B
                  | (1u << 20)                             // pad_enable
                  | ((unsigned)pad_interval_code << 22)
                  | ((unsigned)pad_amount_code << 25));
    g1[1] = (int)((dim0 & 0xFFFFu) << 16);                 // tensor_dim0[15:0]
    g1[2] = (int)((dim0 >> 16) | ((dim1 & 0xFFFFu) << 16));
    g1[3] = (int)((dim1 >> 16) | ((unsigned)tile_w << 16)); // tile_dim0
    g1[4] = (int)(unsigned)tile_h;                          // tile_dim1
    g1[5] = (int)(unsigned)row_stride;                      // tensor_dim0_stride lo
    g1[6] = 0;
    g1[7] = 0;

    v4i gz4;
    gz4[0] = 0; gz4[1] = 0; gz4[2] = 0; gz4[3] = 0;
    v8i gz8;
    for (int q = 0; q < 8; ++q) gz8[q] = 0;

    __builtin_amdgcn_tensor_load_to_lds(g0, g1, gz4, gz4, gz8, 0);
}

// ---------------------------------------------------------------------------
// C = op(A0)*B0 [+ s1 * op(A1)*B1], op(A)=A^T when TRANSA.
// C element stride esC (2 => interleaved complex plane).
// 256 threads = 8 wave32; block tile 128x64, K tile 16; each wave owns 32x32.
// TDM-staged, double-buffered LDS.
// ---------------------------------------------------------------------------
template <bool TRANSA, bool DUAL>
__global__ __launch_bounds__(256)
void gemm_f32_wmma(const float* __restrict__ A0, const float* __restrict__ B0,
                   const float* __restrict__ A1, const float* __restrict__ B1,
                   float s1, float* __restrict__ Cp, int esC) {
    constexpr int NB  = DUAL ? 2 : 1;
    constexpr int AH  = TRANSA ? BK : BM;       // A tile rows (memory order)
    constexpr int AW  = TRANSA ? BM : BK;       // A tile row width (dwords)
    constexpr int AP  = TRANSA ? 136 : 18;      // padded LDS row stride
    constexpr int BP  = 72;                     // B padded row stride
    constexpr int API = TRANSA ? 6 : 3;         // pad_interval code (128 / 16 dw)
    constexpr int APA = TRANSA ? 7 : 1;         // pad_amount code   (8  / 2  dw)
    constexpr short OPS = 2 * NB;               // TDM ops per stage

    __shared__ float sA[2][NB][AH][AP];
    __shared__ float sB[2][NB][BK][BP];

    const int tid  = threadIdx.x;
    const int wave = tid >> 5;
    const int lane = tid & 31;
    const int l    = lane & 15;
    const int hi   = lane >> 4;
    const int wr   = wave >> 1;      // 0..3 -> 32-row stripe
    const int wc   = wave & 1;       // 0..1 -> 32-col stripe
    const int blockM = blockIdx.y * BM;
    const int blockN = blockIdx.x * BN;

    v8f acc[NB][2][2];
    for (int b = 0; b < NB; ++b)
        for (int i = 0; i < 2; ++i)
            for (int j = 0; j < 2; ++j)
                for (int e = 0; e < 8; ++e) acc[b][i][j][e] = 0.f;

    auto issue = [&](int buf, int kt) {
        for (int b = 0; b < NB; ++b) {
            const float* A = b ? A1 : A0;
            const float* B = b ? B1 : B0;
            const float* aptr = TRANSA
                ? (A + (size_t)kt * NMAT + blockM)       // tile [k][m]
                : (A + (size_t)blockM * NMAT + kt);      // tile [m][k]
            tdm_load_2d((unsigned)(uintptr_t)&sA[buf][b][0][0], aptr,
                        AW, AH, NMAT, API, APA);
            tdm_load_2d((unsigned)(uintptr_t)&sB[buf][b][0][0],
                        B + (size_t)kt * NMAT + blockN,
                        BN, BK, NMAT, 5, 7);
        }
    };

    constexpr int T = NMAT / BK;

    if (tid < 32) issue(0, 0);           // prologue: stage buffer 0

    for (int t = 0; t < T; ++t) {
        const int cur = t & 1;
        __syncthreads();                 // everyone done reading buf cur^1
        if (tid < 32) {                  // wave 0: prefetch next, drain current
            if (t + 1 < T) {
                issue(cur ^ 1, (t + 1) * BK);
                __builtin_amdgcn_s_wait_tensorcnt(OPS);
            } else {
                __builtin_amdgcn_s_wait_tensorcnt((short)0);
            }
        }
        __syncthreads();                 // publish buf cur to all waves

        for (int b = 0; b < NB; ++b) {
            for (int kk = 0; kk < BK; kk += 4) {
                // A frag 16x4 f32: lane l -> (M=l, K=kk,kk+1); lane l+16 -> K=kk+2,kk+3
                v2f fa[2];
                for (int i = 0; i < 2; ++i) {
                    int m = wr * 32 + i * 16 + l;
                    if (TRANSA) {
                        fa[i][0] = sA[cur][b][kk + hi * 2 + 0][m];
                        fa[i][1] = sA[cur][b][kk + hi * 2 + 1][m];
                    } else {
                        fa[i][0] = sA[cur][b][m][kk + hi * 2 + 0];
                        fa[i][1] = sA[cur][b][m][kk + hi * 2 + 1];
                    }
                }
                // B frag 4x16 f32: vgpr0 = K rows {kk,kk+2}, vgpr1 = {kk+1,kk+3}
                v2f fb[2];
                for (int j = 0; j < 2; ++j) {
                    int n = wc * 32 + j * 16 + l;
                    fb[j][0] = sB[cur][b][kk + hi * 2 + 0][n];
                    fb[j][1] = sB[cur][b][kk + hi * 2 + 1][n];
                }
                for (int i = 0; i < 2; ++i)
                    for (int j = 0; j < 2; ++j)
                        acc[b][i][j] = __builtin_amdgcn_wmma_f32_16x16x4_f32(
                            false, fa[i], false, fb[j],
                            (short)0, acc[b][i][j], false, false);
            }
        }
    }

    // C/D layout: VGPR r -> M = r + hi*8, N = l
    for (int i = 0; i < 2; ++i) {
        int gmb = blockM + wr * 32 + i * 16 + hi * 8;
        for (int j = 0; j < 2; ++j) {
            int gn = blockN + wc * 32 + j * 16 + l;
            float* cp = Cp + ((size_t)gmb * NMAT + gn) * (size_t)esC;
            for (int r = 0; r < 8; ++r) {
                float v = acc[0][i][j][r];
                if (DUAL) v = fmaf(s1, acc[NB - 1][i][j][r], v);
                cp[(size_t)r * NMAT * esC] = v;
            }
        }
    }
}

extern "C" void kernel_launch(void* const* d_in, const int* in_sizes, int n_in,
                              void* d_out, int out_size, void* d_ws, size_t ws_size,
                              hipStream_t stream) {
    const float* x      = (const float*)d_in[0];   // [4096,4096] f32
    const float* weight = (const float*)d_in[1];   // [15] f32
    float* out = (float*)d_out;                    // complex64 interleaved

    const size_t P = (size_t)NMAT * NMAT;
    float* u2 = (float*)d_ws;          // 32 floats
    float* Fr = (float*)d_ws + 64;     // 256B-aligned planes
    float* Fi = Fr + P;
    float* Tr = Fi + P;
    float* Ti = Tr + P;

    build_u2_kernel<<<1, 32, 0, stream>>>(weight, u2);
    build_F_kernel<<<(unsigned)(P / 256), 256, 0, stream>>>(u2, Fr, Fi);

    dim3 grid(NMAT / BN, NMAT / BM);
    // T = X * F (two real GEMMs)
    gemm_f32_wmma<false, false><<<grid, 256, 0, stream>>>(x, Fr, x, Fr, 0.f, Tr, 1);
    gemm_f32_wmma<false, false><<<grid, 256, 0, stream>>>(x, Fi, x, Fi, 0.f, Ti, 1);
    // outr = Fr^T Tr + Fi^T Ti ; outi = Fr^T Ti - Fi^T Tr
    gemm_f32_wmma<true, true><<<grid, 256, 0, stream>>>(Fr, Tr, Fi, Ti,  1.f, out,     2);
    gemm_f32_wmma<true, true><<<grid, 256, 0, stream>>>(Fr, Ti, Fi, Tr, -1.f, out + 1, 2);
}